// MultiHopNeighborEncoder_45028437131899
// MI455X (gfx1250) — compile-verified
//
#include <hip/hip_runtime.h>
#include <hip/hip_bf16.h>
#include <math.h>

#define NNODES   20000
#define NEDGES   400000
#define IN_C     256
#define HID      128
#define HEADS    2
#define FEAT     256      // HEADS*HID
#define OUT_C    256
#define OUT_T    257      // OUT_C + time
#define SLOPE    0.2f
#define EPSV     1e-16f

typedef __attribute__((ext_vector_type(2))) float v2f;
typedef __attribute__((ext_vector_type(8))) float v8f;

// ---------------------------------------------------------------------------
// helpers
// ---------------------------------------------------------------------------
__device__ __forceinline__ float waveReduceSum(float v) {
    // wave32: full butterfly reduction, all lanes end with the sum
    for (int off = 16; off > 0; off >>= 1) v += __shfl_xor(v, off, 32);
    return v;
}

__device__ __forceinline__ float sigmoidf_(float x) {
    return 1.0f / (1.0f + __expf(-x));
}

// float atomic max via monotonic int mapping (init value must be -inf)
__device__ __forceinline__ void atomicMaxF(float* addr, float val) {
    if (val >= 0.0f) {
        atomicMax((int*)addr, __float_as_int(val));
    } else {
        atomicMin((unsigned int*)addr, (unsigned int)__float_as_int(val));
    }
}

// ---------------------------------------------------------------------------
// WMMA fp32 GEMM:  D[M,256] = A[M,256] @ B[256,256] (+bias_n) (*row_scale)
// one wave = one 32x16 output tile (two 16x16 C accumulators sharing the
// B fragment), K=256 in steps of 4 via V_WMMA_F32_16X16X4_F32.
// Branch-free inner loop; epilogue handles optional bias / row scale.
// ---------------------------------------------------------------------------
__global__ __launch_bounds__(256)
void gemm_wmma_f32(const float* __restrict__ A,
                   const float* __restrict__ B,
                   float* __restrict__ D,
                   int M,
                   const float* __restrict__ bias_n,    // len 256 or null
                   const float* __restrict__ row_scale) // len M or null
{
    const int lane    = threadIdx.x & 31;
    const int waveInB = threadIdx.x >> 5;
    const int wave    = blockIdx.x * 8 + waveInB;

    const int ntiles  = FEAT / 16;            // 16
    const int mtile2  = wave / ntiles;        // 32-row super-tile index
    const int ntile   = wave % ntiles;
    const int m0      = mtile2 * 32;
    const int n0      = ntile * 16;
    if (m0 >= M) return;

    const int lrow  = lane & 15;      // M (for A) / N (for B) index within tile
    const int khalf = lane >> 4;      // which K pair this lane holds

    const float* Arow0 = A + (size_t)(m0 + lrow) * FEAT;
    const float* Arow1 = A + (size_t)(m0 + 16 + lrow) * FEAT;
    const float* Bcol  = B + n0 + lrow;

    v8f c0 = {};
    v8f c1 = {};
    #pragma unroll 4
    for (int kk = 0; kk < FEAT; kk += 4) {
        const int kb = kk + khalf * 2;
        // B fragment: 4x16, row k striped across lanes within VGPR (shared)
        v2f b;
        b.x = Bcol[(size_t)kb * FEAT];
        b.y = Bcol[(size_t)(kb + 1) * FEAT];
        // A fragments: 16x4, row m striped across 2 VGPRs within lane
        v2f a0, a1;
        a0.x = Arow0[kb]; a0.y = Arow0[kb + 1];
        a1.x = Arow1[kb]; a1.y = Arow1[kb + 1];
        c0 = __builtin_amdgcn_wmma_f32_16x16x4_f32(
                false, a0, false, b, (short)0, c0, false, false);
        c1 = __builtin_amdgcn_wmma_f32_16x16x4_f32(
                false, a1, false, b, (short)0, c1, false, false);
    }

    // C/D layout: VGPR v, lanes 0-15 -> M=v, lanes 16-31 -> M=v+8; N = lane&15
    const int col = n0 + lrow;
    const float bn = bias_n ? bias_n[col] : 0.0f;
    #pragma unroll
    for (int v = 0; v < 8; ++v) {
        const int row0 = m0 + v + khalf * 8;
        const int row1 = row0 + 16;
        float v0 = c0[v] + bn;
        float v1 = c1[v] + bn;
        if (row_scale) { v0 *= row_scale[row0]; v1 *= row_scale[row1]; }
        D[(size_t)row0 * FEAT + col] = v0;
        D[(size_t)row1 * FEAT + col] = v1;
    }
}

// ---------------------------------------------------------------------------
// per-(node,head) attention coefficients
// ---------------------------------------------------------------------------
__global__ void alpha_kernel(const float* __restrict__ h,
                             const float* __restrict__ a_src,
                             const float* __restrict__ a_dst,
                             float* __restrict__ as_, float* __restrict__ ad_)
{
    int gid = blockIdx.x * blockDim.x + threadIdx.x;
    if (gid >= NNODES * HEADS) return;
    const int node = gid >> 1;
    const int head = gid & 1;
    const float* hp = h + (size_t)node * FEAT + head * HID;
    const float* ws = a_src + head * HID;
    const float* wd = a_dst + head * HID;
    float ss = 0.0f, sd = 0.0f;
    #pragma unroll 8
    for (int j = 0; j < HID; ++j) {
        const float hv = hp[j];
        ss += hv * ws[j];
        sd += hv * wd[j];
    }
    as_[gid] = ss;
    ad_[gid] = sd;
}

// ---------------------------------------------------------------------------
// per-hop init / edge passes
// ---------------------------------------------------------------------------
__global__ void init_hop_kernel(float* __restrict__ m, float* __restrict__ denom)
{
    int gid = blockIdx.x * blockDim.x + threadIdx.x;
    if (gid >= NNODES * HEADS) return;
    m[gid] = -INFINITY;
    denom[gid] = 0.0f;
}

// seed the GAT accumulator with the broadcast bias bg (so the following GEMM
// needs no A-side bias: (segsum + bg) @ W == GEMM over (acc seeded at bg))
__global__ void acc_init_kernel(float* __restrict__ acc, const float* __restrict__ bg)
{
    int gid = blockIdx.x * blockDim.x + threadIdx.x;
    if (gid >= NNODES * FEAT) return;
    acc[gid] = bg[gid & (FEAT - 1)];
}

__global__ void edge_pass1(const int* __restrict__ src, const int* __restrict__ dst,
                           const float* __restrict__ as_, const float* __restrict__ ad_,
                           float* __restrict__ ew, float* __restrict__ m)
{
    int gid = blockIdx.x * blockDim.x + threadIdx.x;
    if (gid >= NEDGES * HEADS) return;
    const int e = gid >> 1;
    const int head = gid & 1;
    const int s = src[e], d = dst[e];
    float v = as_[s * HEADS + head] + ad_[d * HEADS + head];
    float logit = (v > 0.0f) ? v : SLOPE * v;   // leaky relu
    ew[gid] = logit;
    atomicMaxF(&m[d * HEADS + head], logit);
}

__global__ void fix_m_kernel(float* __restrict__ m)
{
    int gid = blockIdx.x * blockDim.x + threadIdx.x;
    if (gid >= NNODES * HEADS) return;
    float v = m[gid];
    if (!isfinite(v)) m[gid] = 0.0f;
}

__global__ void edge_pass2(const int* __restrict__ dst,
                           const float* __restrict__ m,
                           float* __restrict__ ew, float* __restrict__ denom)
{
    int gid = blockIdx.x * blockDim.x + threadIdx.x;
    if (gid >= NEDGES * HEADS) return;
    const int e = gid >> 1;
    const int head = gid & 1;
    const int d = dst[e];
    float ex = __expf(ew[gid] - m[d * HEADS + head]);
    ew[gid] = ex;
    atomicAdd(&denom[d * HEADS + head], ex);
}

// one thread per (edge, channel): gather h[src], scatter-add into acc[dst]
__global__ __launch_bounds__(256)
void edge_pass3(const int* __restrict__ src, const int* __restrict__ dst,
                const float* __restrict__ h, const float* __restrict__ ew,
                const float* __restrict__ denom, float* __restrict__ acc)
{
    const int e = blockIdx.x;            // 400000 blocks
    const int c = threadIdx.x;           // 256 channels
    const int head = c >> 7;
    const int s = src[e], d = dst[e];
    const float alpha = ew[e * HEADS + head] / (denom[d * HEADS + head] + EPSV);
    const float msg = h[(size_t)s * FEAT + c] * alpha;
    atomicAdd(&acc[(size_t)d * FEAT + c], msg);
}

// ---------------------------------------------------------------------------
// node_time = segment_mean(edge_time_1hop, dst1)
// ---------------------------------------------------------------------------
__global__ void time_accum_kernel(const int* __restrict__ dst,
                                  const float* __restrict__ et,
                                  float* __restrict__ tsum, float* __restrict__ cnt)
{
    int e = blockIdx.x * blockDim.x + threadIdx.x;
    if (e >= NEDGES) return;
    const int d = dst[e];
    atomicAdd(&tsum[d], et[e]);
    atomicAdd(&cnt[d], 1.0f);
}

__global__ void time_mean_kernel(const float* __restrict__ tsum,
                                 const float* __restrict__ cnt,
                                 float* __restrict__ ntime)
{
    int i = blockIdx.x * blockDim.x + threadIdx.x;
    if (i >= NNODES) return;
    ntime[i] = tsum[i] / (cnt[i] + EPSV);
}

// ---------------------------------------------------------------------------
// semantic gate: per 1hop-edge 386-wide dot -> sigmoid -> segment mean on dst
// wave per edge; lanes stride the feature dimension
// ---------------------------------------------------------------------------
__global__ __launch_bounds__(256)
void sem_edge_kernel(const int* __restrict__ src, const int* __restrict__ dst,
                     const float* __restrict__ x1, const float* __restrict__ ntime,
                     const float* __restrict__ rel, const float* __restrict__ et,
                     const float* __restrict__ Wsg, const float* __restrict__ bsg,
                     float* __restrict__ semsum)
{
    const int lane = threadIdx.x & 31;
    const int e = blockIdx.x * 8 + (threadIdx.x >> 5);
    if (e >= NEDGES) return;
    const int s = src[e], d = dst[e];
    const float* x1r  = x1 + (size_t)s * FEAT;
    const float* relr = rel + (size_t)d * HID;
    float sum = 0.0f;
    for (int k = lane; k < OUT_T + HID + 1; k += 32) {   // 386 terms
        float a;
        if (k < OUT_C)             a = x1r[k];            // x_1hop[src]
        else if (k == OUT_C)       a = ntime[s];          // node_time[src]
        else if (k < OUT_T + HID)  a = relr[k - OUT_T];   // rel_embeddings[dst]
        else                       a = et[e];             // edge time
        sum += a * Wsg[k];
    }
    sum = waveReduceSum(sum);
    if (lane == 0) {
        float w = sigmoidf_(sum + bsg[0]);
        atomicAdd(&semsum[d], w);
    }
}

__global__ void sem_mean_kernel(const float* __restrict__ semsum,
                                const float* __restrict__ cnt,
                                float* __restrict__ semw)
{
    int i = blockIdx.x * blockDim.x + threadIdx.x;
    if (i >= NNODES) return;
    semw[i] = semsum[i] / (cnt[i] + EPSV);
}

// ---------------------------------------------------------------------------
// final temporal gate + blend; wave per node
// ---------------------------------------------------------------------------
__global__ __launch_bounds__(256)
void gate_kernel(const float* __restrict__ x1, const float* __restrict__ x2,
                 const float* __restrict__ ntime,
                 const float* __restrict__ Wtg, const float* __restrict__ btg,
                 float* __restrict__ out)
{
    const int lane = threadIdx.x & 31;
    const int i = blockIdx.x * 8 + (threadIdx.x >> 5);
    if (i >= NNODES) return;
    const float* x1r = x1 + (size_t)i * FEAT;
    const float* x2r = x2 + (size_t)i * FEAT;
    const float nt = ntime[i];
    float sum = 0.0f;
    for (int k = lane; k < 2 * OUT_T; k += 32) {         // 514 terms
        float a;
        if (k < OUT_C)             a = x1r[k];
        else if (k == OUT_C)       a = nt;
        else if (k < OUT_T + OUT_C) a = x2r[k - OUT_T];
        else                       a = nt;               // k == 513
        sum += a * Wtg[k];
    }
    sum = waveReduceSum(sum);
    const float g = sigmoidf_(sum + btg[0]);
    float* orow = out + (size_t)i * OUT_T;
    for (int j = lane; j < OUT_T; j += 32) {
        const float a1 = (j < OUT_C) ? x1r[j] : nt;
        const float a2 = (j < OUT_C) ? x2r[j] : nt;
        orow[j] = g * a2 + (1.0f - g) * a1;
    }
}

// ---------------------------------------------------------------------------
// launcher
// ---------------------------------------------------------------------------
extern "C" void kernel_launch(void* const* d_in, const int* in_sizes, int n_in,
                              void* d_out, int out_size, void* d_ws, size_t ws_size,
                              hipStream_t stream)
{
    const float* x     = (const float*)d_in[0];
    const int*   ei1   = (const int*)  d_in[1];
    const float* et1   = (const float*)d_in[2];
    const int*   ei2   = (const int*)  d_in[3];
    const float* et2   = (const float*)d_in[4];
    const float* rel   = (const float*)d_in[5];
    const float* Wg    = (const float*)d_in[6];
    const float* a_src = (const float*)d_in[7];
    const float* a_dst = (const float*)d_in[8];
    const float* bg    = (const float*)d_in[9];
    const float* Wlin  = (const float*)d_in[10];
    const float* blin  = (const float*)d_in[11];
    const float* Wtg   = (const float*)d_in[12];
    const float* btg   = (const float*)d_in[13];
    const float* Wsg   = (const float*)d_in[14];
    const float* bsg   = (const float*)d_in[15];
    float* out = (float*)d_out;

    const int* src1 = ei1;            const int* dst1 = ei1 + NEDGES;
    const int* src2 = ei2;            const int* dst2 = ei2 + NEDGES;

    // workspace partition (floats)
    float* ws = (float*)d_ws;
    float* h      = ws;                       ws += (size_t)NNODES * FEAT;
    float* acc    = ws;                       ws += (size_t)NNODES * FEAT;
    float* x1     = ws;                       ws += (size_t)NNODES * FEAT;
    float* x2     = ws;                       ws += (size_t)NNODES * FEAT;
    float* as_    = ws;                       ws += (size_t)NNODES * HEADS;
    float* ad_    = ws;                       ws += (size_t)NNODES * HEADS;
    float* m      = ws;                       ws += (size_t)NNODES * HEADS;
    float* denom  = ws;                       ws += (size_t)NNODES * HEADS;
    float* ew     = ws;                       ws += (size_t)NEDGES * HEADS;
    float* tsum   = ws;                       ws += NNODES;
    float* cnt    = ws;                       ws += NNODES;
    float* ntime  = ws;                       ws += NNODES;
    float* semsum = ws;                       ws += NNODES;
    float* semw   = ws;                       ws += NNODES;

    const int T = 256;
    const int gemmBlocks = (NNODES / 32) * (FEAT / 16) / 8;   // 1250
    const int nhB  = (NNODES * HEADS + T - 1) / T;
    const int nfB  = (NNODES * FEAT + T - 1) / T;
    const int e2B  = (NEDGES * HEADS + T - 1) / T;
    const int nB   = (NNODES + T - 1) / T;
    const int eB   = (NEDGES + T - 1) / T;

    // h = x @ Wg ; alpha coefficients
    gemm_wmma_f32<<<gemmBlocks, T, 0, stream>>>(x, Wg, h, NNODES, nullptr, nullptr);
    alpha_kernel<<<nhB, T, 0, stream>>>(h, a_src, a_dst, as_, ad_);

    // ---- hop 1 GAT ----
    init_hop_kernel<<<nhB, T, 0, stream>>>(m, denom);
    acc_init_kernel<<<nfB, T, 0, stream>>>(acc, bg);
    edge_pass1<<<e2B, T, 0, stream>>>(src1, dst1, as_, ad_, ew, m);
    fix_m_kernel<<<nhB, T, 0, stream>>>(m);
    edge_pass2<<<e2B, T, 0, stream>>>(dst1, m, ew, denom);
    edge_pass3<<<NEDGES, T, 0, stream>>>(src1, dst1, h, ew, denom, acc);
    gemm_wmma_f32<<<gemmBlocks, T, 0, stream>>>(acc, Wlin, x1, NNODES, blin, nullptr);

    // node_time (segment mean of 1hop edge times)
    hipMemsetAsync(tsum, 0, NNODES * sizeof(float), stream);
    hipMemsetAsync(cnt,  0, NNODES * sizeof(float), stream);
    time_accum_kernel<<<eB, T, 0, stream>>>(dst1, et1, tsum, cnt);
    time_mean_kernel<<<nB, T, 0, stream>>>(tsum, cnt, ntime);

    // semantic gate (uses x1, node_time)
    hipMemsetAsync(semsum, 0, NNODES * sizeof(float), stream);
    sem_edge_kernel<<<(NEDGES + 7) / 8, T, 0, stream>>>(src1, dst1, x1, ntime, rel, et1,
                                                        Wsg, bsg, semsum);
    sem_mean_kernel<<<nB, T, 0, stream>>>(semsum, cnt, semw);

    // ---- hop 2 GAT ----
    init_hop_kernel<<<nhB, T, 0, stream>>>(m, denom);
    acc_init_kernel<<<nfB, T, 0, stream>>>(acc, bg);
    edge_pass1<<<e2B, T, 0, stream>>>(src2, dst2, as_, ad_, ew, m);
    fix_m_kernel<<<nhB, T, 0, stream>>>(m);
    edge_pass2<<<e2B, T, 0, stream>>>(dst2, m, ew, denom);
    edge_pass3<<<NEDGES, T, 0, stream>>>(src2, dst2, h, ew, denom, acc);
    gemm_wmma_f32<<<gemmBlocks, T, 0, stream>>>(acc, Wlin, x2, NNODES, blin, semw);

    // final gate + blend -> (N, 257)
    gate_kernel<<<(NNODES + 7) / 8, T, 0, stream>>>(x1, x2, ntime, Wtg, btg, out);

    (void)in_sizes; (void)n_in; (void)out_size; (void)ws_size; (void)et2;
}